// OHEMLoss_49306224558088
// MI455X (gfx1250) — compile-verified
//
#include <hip/hip_runtime.h>
#include <math.h>

// Problem constants (SHAPE = (16,1,1024,1024), RATIO = 0.25)
#define N_TOTAL (16 * 1024 * 1024)
#define K_TOP   (N_TOTAL / 4)

typedef float v8f __attribute__((ext_vector_type(8)));
typedef float v2f __attribute__((ext_vector_type(2)));

// ---------------- workspace layout ----------------
// [0, N*4)                      : u32 keys (loss bit patterns; loss >= 0 so bits are order-preserving)
// KEYS_BYTES + 0                : hist0[4096]  (key bits [31:20])
// KEYS_BYTES + 16384            : hist1[4096]  (key bits [19:8])
// KEYS_BYTES + 32768            : hist2[256]   (key bits [7:0])
// KEYS_BYTES + 33792            : ctl[16]  (u32): [0]=prefix [1]=k_rem [3]=threshold key T
// KEYS_BYTES + 33856            : double sum accumulator (8B aligned)
#define KEYS_BYTES   ((size_t)N_TOTAL * 4)
#define HIST0_OFF    (KEYS_BYTES)
#define HIST1_OFF    (KEYS_BYTES + 16384)
#define HIST2_OFF    (KEYS_BYTES + 32768)
#define CTL_OFF      (KEYS_BYTES + 33792)
#define SUM_OFF      (KEYS_BYTES + 33856)
#define ZERO_BYTES   (16384 + 16384 + 1024 + 64 + 8)

__device__ __forceinline__ float bce_logits(float x, float t) {
    // max(x,0) - x*t + log1p(exp(-|x|))   (numerically stable, always >= 0)
    return fmaxf(x, 0.0f) - x * t + log1pf(expf(-fabsf(x)));
}

// -------- Pass 1: loss + key store + level-0 histogram (bits [31:20]) --------
__global__ __launch_bounds__(256)
void loss_hist_kernel(const float* __restrict__ pred,
                      const float* __restrict__ targ,
                      unsigned int* __restrict__ keys,
                      unsigned int* __restrict__ hist0) {
    __shared__ unsigned int lh[4096];
    for (int i = threadIdx.x; i < 4096; i += blockDim.x) lh[i] = 0u;
    __syncthreads();

    const int n4     = N_TOTAL / 4;
    const int stride = gridDim.x * blockDim.x;
    const float4* __restrict__ p4 = (const float4*)pred;
    const float4* __restrict__ t4 = (const float4*)targ;
    uint4* __restrict__ k4 = (uint4*)keys;

    for (int i = blockIdx.x * blockDim.x + threadIdx.x; i < n4; i += stride) {
        __builtin_prefetch((const void*)(p4 + i + stride), 0, 0); // global_prefetch_b8
        __builtin_prefetch((const void*)(t4 + i + stride), 0, 0);
        float4 p = p4[i];
        float4 t = t4[i];
        uint4 k;
        k.x = __float_as_uint(bce_logits(p.x, t.x));
        k.y = __float_as_uint(bce_logits(p.y, t.y));
        k.z = __float_as_uint(bce_logits(p.z, t.z));
        k.w = __float_as_uint(bce_logits(p.w, t.w));
        k4[i] = k;
        atomicAdd(&lh[k.x >> 20], 1u);
        atomicAdd(&lh[k.y >> 20], 1u);
        atomicAdd(&lh[k.z >> 20], 1u);
        atomicAdd(&lh[k.w >> 20], 1u);
    }
    __syncthreads();
    for (int i = threadIdx.x; i < 4096; i += blockDim.x)
        if (lh[i]) atomicAdd(&hist0[i], lh[i]);
}

// -------- Select threshold bin of one radix level (1 block, 256 threads) --------
__global__ __launch_bounds__(256)
void select_bin_kernel(const unsigned int* __restrict__ hist,
                       unsigned int* __restrict__ ctl,
                       int nbins, int bits, int level, int is_final) {
    __shared__ unsigned int csum[256];
    const int t = threadIdx.x;
    const int chunk = nbins >> 8;           // 16 (4096 bins) or 1 (256 bins)

    unsigned int local[16];
    unsigned int cs = 0;
    for (int i = 0; i < chunk; ++i) { local[i] = hist[t * chunk + i]; cs += local[i]; }
    csum[t] = cs;
    __syncthreads();
    // inclusive suffix scan of per-thread chunk sums
    for (int off = 1; off < 256; off <<= 1) {
        unsigned int v = csum[t] + ((t + off < 256) ? csum[t + off] : 0u);
        __syncthreads();
        csum[t] = v;
        __syncthreads();
    }
    const unsigned int strictAbove = csum[t] - cs;   // count in chunks strictly above mine
    const unsigned int krem = (level == 0) ? (unsigned int)K_TOP : ctl[1];

    unsigned int run = strictAbove;                  // walk my chunk from the top bin down
    for (int i = chunk - 1; i >= 0; --i) {
        const unsigned int c = local[i];
        if (run < krem && run + c >= krem) {         // exactly one (thread,bin) satisfies this
            const unsigned int bin = (unsigned int)(t * chunk + i);
            unsigned int prefix = (level == 0) ? 0u : ctl[0];
            prefix = (prefix << bits) | bin;
            ctl[0] = prefix;
            ctl[1] = krem - run;                     // elements still needed from inside this bin
            if (is_final) ctl[3] = prefix;           // full 32-bit threshold key T
        }
        run += c;
    }
}

// -------- Refinement histogram: next key bits among prefix-matching keys --------
__global__ __launch_bounds__(256)
void refine_hist_kernel(const unsigned int* __restrict__ keys,
                        const unsigned int* __restrict__ ctl,
                        unsigned int* __restrict__ hist,
                        int matchShift, int binShift, unsigned int binMask, int nbins) {
    __shared__ unsigned int lh[4096];
    for (int i = threadIdx.x; i < nbins; i += blockDim.x) lh[i] = 0u;
    __syncthreads();

    const unsigned int prefix = ctl[0];
    const int n4     = N_TOTAL / 4;
    const int stride = gridDim.x * blockDim.x;
    const uint4* __restrict__ k4 = (const uint4*)keys;

    for (int i = blockIdx.x * blockDim.x + threadIdx.x; i < n4; i += stride) {
        __builtin_prefetch((const void*)(k4 + i + stride), 0, 0);
        uint4 k = k4[i];
        if ((k.x >> matchShift) == prefix) atomicAdd(&lh[(k.x >> binShift) & binMask], 1u);
        if ((k.y >> matchShift) == prefix) atomicAdd(&lh[(k.y >> binShift) & binMask], 1u);
        if ((k.z >> matchShift) == prefix) atomicAdd(&lh[(k.z >> binShift) & binMask], 1u);
        if ((k.w >> matchShift) == prefix) atomicAdd(&lh[(k.w >> binShift) & binMask], 1u);
    }
    __syncthreads();
    for (int i = threadIdx.x; i < nbins; i += blockDim.x)
        if (lh[i]) atomicAdd(&hist[i], lh[i]);
}

// -------- Final sum of values strictly above threshold key T --------
// Wave reduction uses v_wmma_f32_16x16x4_f32: A = (partial, 0) per lane, B = ones.
// D[m][n] = partial[m] + partial[m+16]; summing the 8 D regs per lane gives the
// half-wave sum, one cross-half shuffle completes the 32-lane sum (exact f32).
__global__ __launch_bounds__(256)
void sum_kernel(const unsigned int* __restrict__ keys,
                const unsigned int* __restrict__ ctl,
                double* __restrict__ sumAcc) {
    const unsigned int T = ctl[3];
    const int n4     = N_TOTAL / 4;
    const int stride = gridDim.x * blockDim.x;
    const uint4* __restrict__ k4 = (const uint4*)keys;

    float part = 0.0f;
    for (int i = blockIdx.x * blockDim.x + threadIdx.x; i < n4; i += stride) {
        uint4 k = k4[i];
        if (k.x > T) part += __uint_as_float(k.x);
        if (k.y > T) part += __uint_as_float(k.y);
        if (k.z > T) part += __uint_as_float(k.z);
        if (k.w > T) part += __uint_as_float(k.w);
    }

    // ---- WMMA wave32 reduction (EXEC all-1s here: uniform code path) ----
    v2f a; a.x = part; a.y = 0.0f;
    v2f b; b.x = 1.0f; b.y = 1.0f;
    v8f c = {};
    v8f d = __builtin_amdgcn_wmma_f32_16x16x4_f32(
        /*neg_a=*/false, a, /*neg_b=*/false, b,
        /*c_mod=*/(short)0, c, /*reuse_a=*/false, /*reuse_b=*/false);
    float s = d[0] + d[1] + d[2] + d[3] + d[4] + d[5] + d[6] + d[7]; // half-wave sum
    s += __shfl_xor(s, 16, 32);                                      // full 32-lane sum

    if ((threadIdx.x & 31) == 0)
        atomicAdd(sumAcc, (double)s);   // global_atomic_add_f64
}

// -------- Finalize: mean = (sum_above + ties * T) / K --------
__global__ void finalize_kernel(const unsigned int* __restrict__ ctl,
                                const double* __restrict__ sumAcc,
                                float* __restrict__ out) {
    const double s    = *sumAcc;
    const double ties = (double)ctl[1];
    const double tval = (double)__uint_as_float(ctl[3]);
    out[0] = (float)((s + ties * tval) / (double)K_TOP);
}

extern "C" void kernel_launch(void* const* d_in, const int* in_sizes, int n_in,
                              void* d_out, int out_size, void* d_ws, size_t ws_size,
                              hipStream_t stream) {
    const float* pred = (const float*)d_in[0];
    const float* targ = (const float*)d_in[1];
    float* out = (float*)d_out;

    char* ws = (char*)d_ws;
    unsigned int* keys  = (unsigned int*)(ws);
    unsigned int* hist0 = (unsigned int*)(ws + HIST0_OFF);
    unsigned int* hist1 = (unsigned int*)(ws + HIST1_OFF);
    unsigned int* hist2 = (unsigned int*)(ws + HIST2_OFF);
    unsigned int* ctl   = (unsigned int*)(ws + CTL_OFF);
    double*       sacc  = (double*)(ws + SUM_OFF);

    // zero histograms + control + sum accumulator every call (graph-capture safe)
    hipMemsetAsync(ws + HIST0_OFF, 0, ZERO_BYTES, stream);

    const dim3 blk(256);
    const dim3 grd(4096);

    // Pass 1: loss + keys + level-0 histogram (bits [31:20], 4096 bins)
    loss_hist_kernel<<<grd, blk, 0, stream>>>(pred, targ, keys, hist0);
    select_bin_kernel<<<1, blk, 0, stream>>>(hist0, ctl, 4096, 12, /*level=*/0, /*final=*/0);

    // Level 1: bits [19:8] among keys matching 12-bit prefix
    refine_hist_kernel<<<grd, blk, 0, stream>>>(keys, ctl, hist1, 20, 8, 0xFFFu, 4096);
    select_bin_kernel<<<1, blk, 0, stream>>>(hist1, ctl, 4096, 12, /*level=*/1, /*final=*/0);

    // Level 2: bits [7:0] among keys matching 24-bit prefix -> exact threshold T
    refine_hist_kernel<<<grd, blk, 0, stream>>>(keys, ctl, hist2, 8, 0, 0xFFu, 256);
    select_bin_kernel<<<1, blk, 0, stream>>>(hist2, ctl, 256, 8, /*level=*/2, /*final=*/1);

    // Sum all values strictly above T (WMMA wave reduction + f64 atomic)
    sum_kernel<<<grd, blk, 0, stream>>>(keys, ctl, sacc);

    // mean = (sum_above + ties*T) / K
    finalize_kernel<<<1, 1, 0, stream>>>(ctl, sacc, out);
}